// mul_attention_12910671692371
// MI455X (gfx1250) — compile-verified
//
#include <hip/hip_runtime.h>

typedef __attribute__((ext_vector_type(2))) float v2f;
typedef __attribute__((ext_vector_type(8))) float v8f;

#define N_DIM 64
#define D_DIM 512
#define ROW_PAD 516          // 512 + 4 floats: row stride 2064B -> bank-conflict-free WMMA packing
#define LN_EPS 1e-5f

// Branch-free tanh: prefer gfx1250 hardware V_TANH_F32; otherwise a
// branchless exp-based form (no EXEC manipulation -> safe around WMMA).
__device__ __forceinline__ float fast_tanh(float x) {
#if defined(__AMDGCN__) && __has_builtin(__builtin_amdgcn_tanhf)
    return __builtin_amdgcn_tanhf(x);
#else
    float t = __expf(2.0f * x);          // v_exp_f32, branch-free
    return (t - 1.0f) / (t + 1.0f);
#endif
}

__global__ __launch_bounds__(512, 2)
void fused_ln_tanh_attn_kernel(const float* __restrict__ emb,
                               const float* __restrict__ gamma,
                               const float* __restrict__ beta,
                               const float* __restrict__ Wv,
                               const float* __restrict__ bv,
                               float* __restrict__ out)
{
    __shared__ __align__(16) float e_lds[N_DIM][ROW_PAD];  // normalized tile (row 0 == query)
    __shared__ __align__(16) float g_lds[D_DIM];
    __shared__ __align__(16) float be_lds[D_DIM];
    __shared__ __align__(16) float w_lds[D_DIM];
    __shared__ float psc[4][N_DIM];                        // per-K-chunk partial scores
    __shared__ float alphas[N_DIM];

    const int tid  = threadIdx.x;
    const int lane = tid & 31;
    const int wave = tid >> 5;                 // 0..15 (wave32)
    const size_t base = (size_t)blockIdx.x * N_DIM * D_DIM;
    const float* __restrict__ ebase = emb + base;
    float*       __restrict__ obase = out + base;

    // ---- Phase 0: stage gamma/beta/Wv (512 threads, 1 elem each) ----
    g_lds[tid]  = gamma[tid];
    be_lds[tid] = beta[tid];
    w_lds[tid]  = Wv[tid];

    // ---- Phase 1: LayerNorm, 4 rows per wave, 16 f32 per lane per row ----
    for (int r = 0; r < 4; ++r) {
        const int row = wave * 4 + r;
        const float* rp = ebase + row * D_DIM;
        float4 x[4];
        float s = 0.f;
        #pragma unroll
        for (int p = 0; p < 4; ++p) {                       // coalesced b128 loads
            x[p] = *(const float4*)(rp + p * 128 + lane * 4);
            s += x[p].x + x[p].y + x[p].z + x[p].w;
        }
        #pragma unroll
        for (int off = 16; off; off >>= 1) s += __shfl_xor(s, off, 32);
        const float mean = s * (1.0f / 512.0f);
        float v = 0.f;
        #pragma unroll
        for (int p = 0; p < 4; ++p) {
            float dx = x[p].x - mean, dy = x[p].y - mean,
                  dz = x[p].z - mean, dw = x[p].w - mean;
            v += dx*dx + dy*dy + dz*dz + dw*dw;
        }
        #pragma unroll
        for (int off = 16; off; off >>= 1) v += __shfl_xor(v, off, 32);
        const float inv = rsqrtf(v * (1.0f / 512.0f) + LN_EPS);
        #pragma unroll
        for (int p = 0; p < 4; ++p) {
            const int idx = p * 128 + lane * 4;
            float4 g4 = *(const float4*)&g_lds[idx];
            float4 b4 = *(const float4*)&be_lds[idx];
            float4 e4;
            e4.x = (x[p].x - mean) * inv * g4.x + b4.x;
            e4.y = (x[p].y - mean) * inv * g4.y + b4.y;
            e4.z = (x[p].z - mean) * inv * g4.z + b4.z;
            e4.w = (x[p].w - mean) * inv * g4.w + b4.w;
            *(float4*)&e_lds[row][idx] = e4;
        }
    }
    __syncthreads();

    // ---- Phase 2: scores via V_WMMA_F32_16X16X4_F32 ----
    // wave -> (row group = wave&3, K chunk = wave>>2); 32 WMMAs per wave.
    // A[m][k] = tanh(q[k]*e[m][k]) (16x4), B[k][n] = Wv[k] broadcast over n (4x16).
    // D[m][*] accumulates the dot product; any consistent K permutation in the
    // A/B lane layouts still produces the exact sum. Loop body is branch-free
    // (hardware v_tanh), so EXEC stays all-1s as WMMA requires.
    {
        const int group = wave & 3;
        const int kbase = (wave >> 2) * 128;
        const int m     = lane & 15;
        const int row   = group * 16 + m;
        const int koff  = (lane < 16) ? 0 : 2;
        v8f acc = {};
        #pragma unroll 4
        for (int k0 = kbase; k0 < kbase + 128; k0 += 4) {
            const int kk = k0 + koff;                       // even -> 8B-aligned b64 loads
            v2f q2 = *(const v2f*)&e_lds[0][kk];
            v2f e2 = *(const v2f*)&e_lds[row][kk];
            v2f w2 = *(const v2f*)&w_lds[kk];
            v2f a;
            a[0] = fast_tanh(q2[0] * e2[0]);
            a[1] = fast_tanh(q2[1] * e2[1]);
            acc = __builtin_amdgcn_wmma_f32_16x16x4_f32(
                /*neg_a=*/false, a, /*neg_b=*/false, w2,
                /*c_mod=*/(short)0, acc, /*reuse_a=*/false, /*reuse_b=*/false);
        }
        // D layout: lane(col) 0..15 -> M = vgpr r; lanes 16..31 -> M = r+8.
        // All N columns identical (B broadcast), so read col 0 / col 16.
        if (lane == 0) {
            #pragma unroll
            for (int r = 0; r < 8; ++r) psc[wave >> 2][group * 16 + r] = acc[r];
        } else if (lane == 16) {
            #pragma unroll
            for (int r = 0; r < 8; ++r) psc[wave >> 2][group * 16 + 8 + r] = acc[r];
        }
    }
    __syncthreads();

    // ---- Phase 3: softmax over scores[1..63] (wave 0), fixed summation order ----
    if (wave == 0) {
        const float bvv = bv[0];
        float s1 = bvv + psc[0][lane]      + psc[1][lane]      + psc[2][lane]      + psc[3][lane];
        float s2 = bvv + psc[0][lane + 32] + psc[1][lane + 32] + psc[2][lane + 32] + psc[3][lane + 32];
        float m1 = (lane >= 1) ? s1 : -3.402823466e38f;       // row 0 is the query, excluded
        float mx = fmaxf(m1, s2);
        #pragma unroll
        for (int off = 16; off; off >>= 1) mx = fmaxf(mx, __shfl_xor(mx, off, 32));
        float e1 = (lane >= 1) ? __expf(s1 - mx) : 0.f;       // branch-free v_exp_f32
        float e2 = __expf(s2 - mx);
        float sum = e1 + e2;
        #pragma unroll
        for (int off = 16; off; off >>= 1) sum += __shfl_xor(sum, off, 32);
        const float rs = 1.0f / sum;
        alphas[lane]      = (lane >= 1) ? e1 * rs : 1.0f;     // alpha[0]=1 -> query passthrough
        alphas[lane + 32] = e2 * rs;
    }
    __syncthreads();

    // ---- Phase 4: scaled write-out, coalesced b128 stores ----
    for (int r = 0; r < 4; ++r) {
        const int row = wave * 4 + r;
        const float al = alphas[row];
        float* op = obase + row * D_DIM;
        #pragma unroll
        for (int p = 0; p < 4; ++p) {
            const int idx = p * 128 + lane * 4;
            float4 e4 = *(const float4*)&e_lds[row][idx];
            float4 o4;
            o4.x = al * e4.x; o4.y = al * e4.y; o4.z = al * e4.z; o4.w = al * e4.w;
            *(float4*)(op + idx) = o4;
        }
    }
}

extern "C" void kernel_launch(void* const* d_in, const int* in_sizes, int n_in,
                              void* d_out, int out_size, void* d_ws, size_t ws_size,
                              hipStream_t stream) {
    const float* emb   = (const float*)d_in[0];
    const float* gamma = (const float*)d_in[1];
    const float* beta  = (const float*)d_in[2];
    const float* Wv    = (const float*)d_in[3];
    const float* bv    = (const float*)d_in[4];
    float* out = (float*)d_out;
    fused_ln_tanh_attn_kernel<<<dim3(4096), dim3(512), 0, stream>>>(
        emb, gamma, beta, Wv, bv, out);
}